// GNN_18837726560924
// MI455X (gfx1250) — compile-verified
//
#include <hip/hip_runtime.h>

#define NN      50000
#define NE      800000
#define EMB     64
#define E2      128
#define LAYERS  5
#define EPSV    1e-5f
#define STRIPS  5            // 16-row strips per block in the GEMM kernels
#define APAD    132          // LDS row stride (floats) for A strips: conflict-free

typedef __attribute__((ext_vector_type(2))) float v2f;
typedef __attribute__((ext_vector_type(8))) float v8f;

// Async bulk copy global -> LDS (CDNA5, ASYNCcnt-tracked, no VGPR staging).
__device__ __forceinline__ void async_g2l_b128(void* lds_dst, const void* gsrc) {
    unsigned           loff = (unsigned)(unsigned long long)lds_dst;  // low 32 = LDS offset
    unsigned long long ga   = (unsigned long long)gsrc;
    asm volatile("global_load_async_to_lds_b128 %0, %1, off"
                 :: "v"(loff), "v"(ga) : "memory");
}
__device__ __forceinline__ void async_wait0() {
    asm volatile("s_wait_asynccnt 0x0" ::: "memory");
}

// ---------------------------------------------------------------------------
// h0[i][j] = node_emb[x[i]][j]
// ---------------------------------------------------------------------------
__global__ __launch_bounds__(256) void k_init_h(const int* __restrict__ x,
                                                const float* __restrict__ node_emb,
                                                float* __restrict__ h) {
    int idx = blockIdx.x * blockDim.x + threadIdx.x;
    if (idx < NN * EMB) {
        int i = idx >> 6, j = idx & 63;
        h[idx] = node_emb[x[i] * EMB + j];
    }
}

// ---------------------------------------------------------------------------
// Self-loop init of agg + zero the BN-stats accumulators.
// agg[i][0:64] = h[i];  agg[i][64:128] = We_l[0,:] + be_l  (attr = [1,0,0])
// ---------------------------------------------------------------------------
__global__ __launch_bounds__(256) void k_init_agg(const float* __restrict__ h,
                                                  const float* __restrict__ We_l,
                                                  const float* __restrict__ be_l,
                                                  float* __restrict__ agg,
                                                  float* __restrict__ stats) {
    int idx = blockIdx.x * blockDim.x + threadIdx.x;
    if (blockIdx.x == 0 && threadIdx.x < 2 * E2) stats[threadIdx.x] = 0.0f;
    if (idx < NN * E2) {
        int i = idx >> 7, j = idx & 127;
        float v;
        if (j < EMB) v = h[i * EMB + j];
        else         v = We_l[j - EMB] + be_l[j - EMB];
        agg[idx] = v;
    }
}

// ---------------------------------------------------------------------------
// Edge scatter: one wave per edge (grid-stride). Lane L handles features
// j = 2L, 2L+1 (coalesced b64 gather of h[src]).
// agg[dst] += [ h[src] , ea@We + be ]  via f32 atomics (L2-resident RMW).
// ---------------------------------------------------------------------------
__global__ __launch_bounds__(256) void k_edges(const int* __restrict__ ei,
                                               const float* __restrict__ ea,
                                               const float* __restrict__ h,
                                               const float* __restrict__ We_l,
                                               const float* __restrict__ be_l,
                                               float* __restrict__ agg) {
    int lane   = threadIdx.x & 31;
    int wave   = blockIdx.x * (blockDim.x >> 5) + (threadIdx.x >> 5);
    int nwaves = gridDim.x * (blockDim.x >> 5);
    int j0 = lane * 2, j1 = lane * 2 + 1;
    float w00 = We_l[0 * EMB + j0], w10 = We_l[1 * EMB + j0], w20 = We_l[2 * EMB + j0];
    float w01 = We_l[0 * EMB + j1], w11 = We_l[1 * EMB + j1], w21 = We_l[2 * EMB + j1];
    float bb0 = be_l[j0], bb1 = be_l[j1];
    for (int e = wave; e < NE; e += nwaves) {
        int s = ei[e];
        int d = ei[NE + e];
        float a0 = ea[e * 3 + 0], a1 = ea[e * 3 + 1], a2 = ea[e * 3 + 2];
        v2f hv = *(const v2f*)(h + (size_t)s * EMB + j0);
        float em0 = fmaf(a0, w00, fmaf(a1, w10, fmaf(a2, w20, bb0)));
        float em1 = fmaf(a0, w01, fmaf(a1, w11, fmaf(a2, w21, bb1)));
        float* base = agg + (size_t)d * E2;
        atomicAdd(base + j0, hv.x);
        atomicAdd(base + j1, hv.y);
        atomicAdd(base + EMB + j0, em0);
        atomicAdd(base + EMB + j1, em1);
    }
}

// ---------------------------------------------------------------------------
// GEMM1: z[N,128] = agg[N,128] @ W1[128,128] + b1, fp32 WMMA 16x16x4.
// Block = 256 thr = 8 waves; block owns STRIPS 16-row strips. W1 staged once
// per block via async global->LDS; A strips double-buffered through LDS with
// async DMA prefetch overlapping the WMMA of the current strip.
// Also accumulates per-feature sum / sum-of-squares for BN (one atomic/lane).
// ---------------------------------------------------------------------------
__global__ __launch_bounds__(256) void k_gemm1(const float* __restrict__ agg,
                                               const float* __restrict__ W1_l,
                                               const float* __restrict__ b1_l,
                                               float* __restrict__ z,
                                               float* __restrict__ stats) {
    __shared__ float lw[E2 * E2];          // 64 KB weights
    __shared__ float la[2][16 * APAD];     // ~16.5 KB A double-buffer
    int tid = threadIdx.x;

    // Stage weights + strip 0 asynchronously.
    for (int i = tid * 4; i < E2 * E2; i += 256 * 4)
        async_g2l_b128(&lw[i], &W1_l[i]);
    {
        int row0 = blockIdx.x * STRIPS * 16;
        for (int c = tid; c < 512; c += 256) {               // 16 rows x 32 chunks
            int row = c >> 5, off = (c & 31) * 4;
            async_g2l_b128(&la[0][row * APAD + off],
                           agg + (size_t)(row0 + row) * E2 + off);
        }
    }
    async_wait0();
    __syncthreads();

    int lane = tid & 31, wave = tid >> 5;
    int m = lane & 15, hi = lane >> 4;
    int col = wave * 16 + m;
    float bias = b1_l[col];

    float s = 0.0f, sq = 0.0f;
#pragma unroll 1
    for (int strip = 0; strip < STRIPS; strip++) {
        int row0 = (blockIdx.x * STRIPS + strip) * 16;
        int sb = strip & 1;
        if (strip + 1 < STRIPS) {                            // prefetch next strip
            int nrow0 = row0 + 16;
            for (int c = tid; c < 512; c += 256) {
                int row = c >> 5, off = (c & 31) * 4;
                async_g2l_b128(&la[sb ^ 1][row * APAD + off],
                               agg + (size_t)(nrow0 + row) * E2 + off);
            }
        }

        const float* arow = &la[sb][m * APAD];
        v8f c = {};
#pragma unroll
        for (int k = 0; k < E2; k += 4) {
            int ka = k + 2 * hi;
            v2f a = *(const v2f*)(arow + ka);       // A[m][ka], A[m][ka+1] (LDS)
            v2f b;
            b.x = lw[ka * E2 + col];                // B[ka][n]  (reg-hoisted)
            b.y = lw[(ka + 1) * E2 + col];
            c = __builtin_amdgcn_wmma_f32_16x16x4_f32(false, a, false, b,
                                                      (short)0, c, false, false);
        }
#pragma unroll
        for (int r = 0; r < 8; r++) {
            float v = c[r] + bias;
            z[(size_t)(row0 + r + 8 * hi) * E2 + col] = v;
            s += v; sq += v * v;
        }
        if (strip + 1 < STRIPS) {
            async_wait0();
            __syncthreads();
        }
    }
    atomicAdd(&stats[col], s);
    atomicAdd(&stats[E2 + col], sq);
}

// ---------------------------------------------------------------------------
// Fold BN stats into an affine transform: zn = z*bnA + bnB
// ---------------------------------------------------------------------------
__global__ __launch_bounds__(128) void k_bnstats(const float* __restrict__ stats,
                                                 const float* __restrict__ gamma_l,
                                                 const float* __restrict__ beta_l,
                                                 float* __restrict__ bnA,
                                                 float* __restrict__ bnB) {
    int j = threadIdx.x;  // 128
    float invN = 1.0f / (float)NN;
    float mu  = stats[j] * invN;
    float var = stats[E2 + j] * invN - mu * mu;
    float inv = rsqrtf(var + EPSV);
    float a = inv * gamma_l[j];
    bnA[j] = a;
    bnB[j] = beta_l[j] - mu * a;
}

// ---------------------------------------------------------------------------
// GEMM2: hout[N,64] = act( relu(z*bnA + bnB) @ W2[128,64] + b2 )
// Block = 128 thr = 4 waves; same async double-buffered A pipeline as GEMM1;
// BN+ReLU fused into the A-operand read from LDS.
// ---------------------------------------------------------------------------
__global__ __launch_bounds__(128) void k_gemm2(const float* __restrict__ z,
                                               const float* __restrict__ bnA,
                                               const float* __restrict__ bnB,
                                               const float* __restrict__ W2_l,
                                               const float* __restrict__ b2_l,
                                               float* __restrict__ hout,
                                               int relu_out) {
    __shared__ float lw[E2 * EMB];         // 32 KB weights
    __shared__ float lz[2][16 * APAD];     // ~16.5 KB A double-buffer
    __shared__ float lA[E2];
    __shared__ float lB[E2];
    int tid = threadIdx.x;

    for (int i = tid * 4; i < E2 * EMB; i += 128 * 4)
        async_g2l_b128(&lw[i], &W2_l[i]);
    {
        int row0 = blockIdx.x * STRIPS * 16;
        for (int c = tid; c < 512; c += 128) {
            int row = c >> 5, off = (c & 31) * 4;
            async_g2l_b128(&lz[0][row * APAD + off],
                           z + (size_t)(row0 + row) * E2 + off);
        }
    }
    if (tid < E2) { lA[tid] = bnA[tid]; lB[tid] = bnB[tid]; }
    async_wait0();
    __syncthreads();

    int lane = tid & 31, wave = tid >> 5;
    int m = lane & 15, hi = lane >> 4;
    int col = wave * 16 + m;
    float bias = b2_l[col];

#pragma unroll 1
    for (int strip = 0; strip < STRIPS; strip++) {
        int row0 = (blockIdx.x * STRIPS + strip) * 16;
        int sb = strip & 1;
        if (strip + 1 < STRIPS) {
            int nrow0 = row0 + 16;
            for (int c = tid; c < 512; c += 128) {
                int row = c >> 5, off = (c & 31) * 4;
                async_g2l_b128(&lz[sb ^ 1][row * APAD + off],
                               z + (size_t)(nrow0 + row) * E2 + off);
            }
        }

        const float* zrow = &lz[sb][m * APAD];
        v8f c = {};
#pragma unroll
        for (int k = 0; k < E2; k += 4) {
            int ka = k + 2 * hi;
            v2f zr = *(const v2f*)(zrow + ka);
            v2f a;
            a.x = fmaxf(fmaf(zr.x, lA[ka],     lB[ka]),     0.0f);
            a.y = fmaxf(fmaf(zr.y, lA[ka + 1], lB[ka + 1]), 0.0f);
            v2f b;
            b.x = lw[ka * EMB + col];
            b.y = lw[(ka + 1) * EMB + col];
            c = __builtin_amdgcn_wmma_f32_16x16x4_f32(false, a, false, b,
                                                      (short)0, c, false, false);
        }
#pragma unroll
        for (int r = 0; r < 8; r++) {
            float v = c[r] + bias;
            if (relu_out) v = fmaxf(v, 0.0f);
            hout[(size_t)(row0 + r + 8 * hi) * EMB + col] = v;
        }
        if (strip + 1 < STRIPS) {
            async_wait0();
            __syncthreads();
        }
    }
}

// ---------------------------------------------------------------------------
extern "C" void kernel_launch(void* const* d_in, const int* in_sizes, int n_in,
                              void* d_out, int out_size, void* d_ws, size_t ws_size,
                              hipStream_t stream) {
    const int*   x        = (const int*)  d_in[0];
    const int*   ei       = (const int*)  d_in[1];
    const float* ea       = (const float*)d_in[2];
    const float* node_emb = (const float*)d_in[3];
    const float* We       = (const float*)d_in[4];
    const float* be       = (const float*)d_in[5];
    const float* W1       = (const float*)d_in[6];
    const float* b1       = (const float*)d_in[7];
    const float* gamma    = (const float*)d_in[8];
    const float* beta     = (const float*)d_in[9];
    const float* W2       = (const float*)d_in[10];
    const float* b2       = (const float*)d_in[11];
    float* out = (float*)d_out;

    // Workspace layout (floats): ~73.3 MiB total — resident in the 192 MB L2.
    float* ws    = (float*)d_ws;
    float* h_a   = ws;                        // N*64
    float* h_b   = h_a + (size_t)NN * EMB;    // N*64
    float* agg   = h_b + (size_t)NN * EMB;    // N*128
    float* zbuf  = agg + (size_t)NN * E2;     // N*128
    float* stats = zbuf + (size_t)NN * E2;    // 2*128
    float* bnA   = stats + 2 * E2;            // 128
    float* bnB   = bnA + E2;                  // 128
    (void)ws_size; (void)n_in; (void)in_sizes; (void)out_size;

    // h0 = node_emb[x]
    k_init_h<<<(NN * EMB + 255) / 256, 256, 0, stream>>>(x, node_emb, h_a);

    float* hin = h_a;
    for (int l = 0; l < LAYERS; l++) {
        const float* We_l = We + (size_t)l * 3 * EMB;
        const float* be_l = be + (size_t)l * EMB;
        const float* W1_l = W1 + (size_t)l * E2 * E2;
        const float* b1_l = b1 + (size_t)l * E2;
        const float* ga_l = gamma + (size_t)l * E2;
        const float* bt_l = beta  + (size_t)l * E2;
        const float* W2_l = W2 + (size_t)l * E2 * EMB;
        const float* b2_l = b2 + (size_t)l * EMB;

        int last = (l == LAYERS - 1);
        float* hout = last ? out : (hin == h_a ? h_b : h_a);

        k_init_agg<<<(NN * E2 + 255) / 256, 256, 0, stream>>>(hin, We_l, be_l, agg, stats);
        k_edges<<<1024, 256, 0, stream>>>(ei, ea, hin, We_l, be_l, agg);
        k_gemm1<<<NN / (16 * STRIPS), 256, 0, stream>>>(agg, W1_l, b1_l, zbuf, stats);
        k_bnstats<<<1, 128, 0, stream>>>(stats, ga_l, bt_l, bnA, bnB);
        k_gemm2<<<NN / (16 * STRIPS), 128, 0, stream>>>(zbuf, bnA, bnB, W2_l, b2_l,
                                                        hout, last ? 0 : 1);
        hin = hout;
    }
}